// ConvertUAVidMasks_70188355551350
// MI455X (gfx1250) — compile-verified
//
#include <hip/hip_runtime.h>
#include <hip/hip_bf16.h>

// Clang native vector types: required by __builtin_nontemporal_load/store
// (HIP_vector_type structs are rejected), and they still lower to b128 VMEM ops.
typedef int   v4i __attribute__((ext_vector_type(4)));
typedef float v4f __attribute__((ext_vector_type(4)));

// UAVid palette keys (r<<16 | g<<8 | b). Class 0 key is 0x000000 and
// contributes 0, so only classes 1..7 need explicit matching; any
// off-palette pixel falls through to 0 automatically.
__device__ __forceinline__ float uavid_classify(unsigned r, unsigned g, unsigned b) {
    unsigned key = (r << 16) | (g << 8) | b;
    unsigned c = 0u;
    c |= (key == 0x800000u) ? 1u : 0u;  // [128,   0,   0]
    c |= (key == 0x804080u) ? 2u : 0u;  // [128,  64, 128]
    c |= (key == 0xC000C0u) ? 3u : 0u;  // [192,   0, 192]
    c |= (key == 0x008000u) ? 4u : 0u;  // [  0, 128,   0]
    c |= (key == 0x808000u) ? 5u : 0u;  // [128, 128,   0]
    c |= (key == 0x404000u) ? 6u : 0u;  // [ 64,  64,   0]
    c |= (key == 0x400080u) ? 7u : 0u;  // [ 64,   0, 128]
    return (float)c;
}

// Main path: 4 pixels per lane via b128 non-temporal loads from the three CHW
// planes and a b128 non-temporal store. Memory-bound streaming:
// 16 B per pixel -> ~5.7 us at 23.3 TB/s.
__global__ __launch_bounds__(256) void ConvertUAVidMasks_vec4_kernel(
    const int* __restrict__ t, float* __restrict__ out, int n4, int plane) {
    const v4i* __restrict__ R = reinterpret_cast<const v4i*>(t);
    const v4i* __restrict__ G = reinterpret_cast<const v4i*>(t + (size_t)plane);
    const v4i* __restrict__ B = reinterpret_cast<const v4i*>(t + 2 * (size_t)plane);
    v4f* __restrict__ O = reinterpret_cast<v4f*>(out);

    int stride = gridDim.x * blockDim.x;
    for (int i = blockIdx.x * blockDim.x + threadIdx.x; i < n4; i += stride) {
        // Non-temporal (TH=NT) b128 loads: single-pass stream, don't pollute L2.
        v4i r = __builtin_nontemporal_load(R + i);
        v4i g = __builtin_nontemporal_load(G + i);
        v4i b = __builtin_nontemporal_load(B + i);

        v4f o;
        o.x = uavid_classify((unsigned)r.x, (unsigned)g.x, (unsigned)b.x);
        o.y = uavid_classify((unsigned)r.y, (unsigned)g.y, (unsigned)b.y);
        o.z = uavid_classify((unsigned)r.z, (unsigned)g.z, (unsigned)b.z);
        o.w = uavid_classify((unsigned)r.w, (unsigned)g.w, (unsigned)b.w);

        __builtin_nontemporal_store(o, O + i);
    }
}

// Scalar tail for plane % 4 != 0 (not hit for 2160x3840, kept for safety).
__global__ __launch_bounds__(256) void ConvertUAVidMasks_tail_kernel(
    const int* __restrict__ t, float* __restrict__ out, int plane, int start) {
    int i = start + blockIdx.x * blockDim.x + threadIdx.x;
    if (i < plane) {
        out[i] = uavid_classify((unsigned)t[i],
                                (unsigned)t[(size_t)plane + i],
                                (unsigned)t[2 * (size_t)plane + i]);
    }
}

extern "C" void kernel_launch(void* const* d_in, const int* in_sizes, int n_in,
                              void* d_out, int out_size, void* d_ws, size_t ws_size,
                              hipStream_t stream) {
    (void)n_in; (void)d_ws; (void)ws_size; (void)out_size;

    const int* tensor = (const int*)d_in[0];   // [3, H, W] int32, CHW planes
    float* out = (float*)d_out;                // [H*W] class ids as float

    int total = in_sizes[0];                   // 3 * H * W
    int plane = total / 3;                     // H * W pixels
    int n4 = plane >> 2;                       // int4 groups
    int tail = plane - (n4 << 2);

    if (n4 > 0) {
        int threads = 256;
        int blocks = (n4 + threads - 1) / threads;   // one v4i per thread
        ConvertUAVidMasks_vec4_kernel<<<blocks, threads, 0, stream>>>(
            tensor, out, n4, plane);
    }
    if (tail > 0) {
        ConvertUAVidMasks_tail_kernel<<<1, 256, 0, stream>>>(
            tensor, out, plane, n4 << 2);
    }
}